// QIntSoftmax_26173530702396
// MI455X (gfx1250) — compile-verified
//
#include <hip/hip_runtime.h>
#include <stdint.h>

typedef __attribute__((ext_vector_type(4))) float f4;

#define ROW 1024
#define TPB 256

// ---- gfx1250 async global->LDS path (guarded; fallback = direct loads) ----
#if defined(__has_builtin)
#  if __has_builtin(__builtin_amdgcn_global_load_async_to_lds_b128) && \
      __has_builtin(__builtin_amdgcn_s_wait_asynccnt)
#    define HAVE_ASYNC_LDS 1
#  endif
#endif
#ifndef HAVE_ASYNC_LDS
#  define HAVE_ASYNC_LDS 0
#endif

#if HAVE_ASYNC_LDS
// Builtin signature (from clang diagnostic): param0 = v4i AS1* (global src),
// param1 = v4i AS3* (LDS dst), then imm offset, imm cpol.
typedef int v4i __attribute__((__vector_size__(16)));
__device__ __forceinline__ void async_copy_b128(void* lds_dst, const void* gsrc) {
  // Build address-space-qualified pointers via integer detour (clang forbids
  // direct generic->AS casts). AS3 pointers are 32-bit: low 32 bits of a
  // generic LDS pointer are the LDS byte offset.
  auto g = (__attribute__((address_space(1))) v4i*)(unsigned long long)gsrc;
  auto l = (__attribute__((address_space(3))) v4i*)(unsigned)(unsigned long long)lds_dst;
  __builtin_amdgcn_global_load_async_to_lds_b128(g, l, 0, 0);
}
#endif

// ---- order-preserving float <-> uint encoding for atomic min/max ----
__device__ __forceinline__ unsigned f32_sortable(float f) {
  unsigned u = __float_as_uint(f);
  return (u & 0x80000000u) ? ~u : (u | 0x80000000u);
}
__device__ __forceinline__ float sortable_f32(unsigned u) {
  unsigned v = (u & 0x80000000u) ? (u & 0x7FFFFFFFu) : ~u;
  return __uint_as_float(v);
}

__global__ void init_gmm(unsigned* gmm) {
  gmm[0] = 0u;           // running sortable max  (smallest possible key)
  gmm[1] = 0xFFFFFFFFu;  // running sortable min  (largest possible key)
}

// Pass 1: per-row max + global min/max. One row (4KB) per 256-thread block.
__global__ void __launch_bounds__(TPB) pass1_rowmax_minmax(
    const float* __restrict__ x, unsigned* __restrict__ gmm,
    float* __restrict__ rowmax) {
  const int row = blockIdx.x;
  const int t   = threadIdx.x;
  const float* src = x + (size_t)row * ROW;

  f4 v;
#if HAVE_ASYNC_LDS
  __shared__ float tile[ROW];
  async_copy_b128(&tile[t * 4], src + t * 4);
  __builtin_amdgcn_s_wait_asynccnt(0);  // this wave's copies landed in LDS
  __syncthreads();                      // all waves' copies landed
  v = *(const f4*)&tile[t * 4];
#else
  v = ((const f4*)src)[t];
#endif

  float mx = fmaxf(fmaxf(v.x, v.y), fmaxf(v.z, v.w));
  float mn = fminf(fminf(v.x, v.y), fminf(v.z, v.w));

  // wave32 butterfly reduction
  #pragma unroll
  for (int off = 16; off >= 1; off >>= 1) {
    mx = fmaxf(mx, __shfl_xor(mx, off, 32));
    mn = fminf(mn, __shfl_xor(mn, off, 32));
  }

  __shared__ float wmx[TPB / 32], wmn[TPB / 32];
  const int wave = t >> 5, lane = t & 31;
  if (lane == 0) { wmx[wave] = mx; wmn[wave] = mn; }
  __syncthreads();
  if (t == 0) {
    float bm = wmx[0], bn = wmn[0];
    #pragma unroll
    for (int i = 1; i < TPB / 32; ++i) {
      bm = fmaxf(bm, wmx[i]);
      bn = fminf(bn, wmn[i]);
    }
    rowmax[row] = bm;
    atomicMax(&gmm[0], f32_sortable(bm));
    atomicMin(&gmm[1], f32_sortable(bn));
  }
}

// Pass 2: integer-exp softmax numerator, streamed with NT stores.
__global__ void __launch_bounds__(TPB) pass2_intexp(
    const float* __restrict__ x, const unsigned* __restrict__ gmm,
    const float* __restrict__ rowmax, float* __restrict__ out) {
#pragma clang fp contract(off)
  const int row = blockIdx.x;
  const int t   = threadIdx.x;

  const float gmax  = sortable_f32(gmm[0]);
  const float gmin  = sortable_f32(gmm[1]);
  float scale = fmaxf((gmax - gmin) / 255.0f, 1e-8f);

  const float s2  = scale * scale;
  const float x0i = floorf(-0.69314718f / scale);                       // floor(X0/scale)
  const float bi  = floorf((float)(0.96963238 / 0.35815147) / scale);   // floor(C1/scale)
  const float ci  = floorf((float)(1.0 / 0.35815147) / s2);             // floor(C2/scale^2)
  const float clampv   = 15.0f * x0i;                                   // N_ITER * x0_int
  const float rmax_int = rowmax[row] / scale;                           // max(x_int) per row
  const float esf = (0.35815147f * s2) / 32768.0f;                      // C0*scale^2 / 2^15

  const float* src = x   + (size_t)row * ROW;
  float*       dst = out + (size_t)row * ROW;

  f4 v = ((const f4*)src)[t];
  f4 o;
  #pragma unroll
  for (int k = 0; k < 4; ++k) {
    float xi  = v[k] / scale;        // element-wise division, like reference
    float xin = xi - rmax_int;       // subtract row max of x_int
    xin = fmaxf(xin, clampv);
    float q = floorf(xin / x0i);
    float r = xin - x0i * q;
    float z = r * (r + bi) + ci;
    int   sh = 15 - (int)q;          // q in [0,15]
    float e  = floorf(ldexpf(z, sh)); // exact z * 2^(15-q), then floor
    e = fmaxf(e, 0.0f);
    o[k] = e * esf;
  }
  __builtin_nontemporal_store(o, (f4*)dst + t);
}

extern "C" void kernel_launch(void* const* d_in, const int* in_sizes, int n_in,
                              void* d_out, int out_size, void* d_ws, size_t ws_size,
                              hipStream_t stream) {
  (void)n_in; (void)out_size; (void)ws_size;
  const float* x = (const float*)d_in[0];
  float* out = (float*)d_out;
  const int n    = in_sizes[0];
  const int rows = n / ROW;  // 4*12*1024 = 49152

  unsigned* gmm    = (unsigned*)d_ws;
  float*    rowmax = (float*)((char*)d_ws + 16);

  init_gmm<<<1, 1, 0, stream>>>(gmm);
  pass1_rowmax_minmax<<<rows, TPB, 0, stream>>>(x, gmm, rowmax);
  pass2_intexp<<<rows, TPB, 0, stream>>>(x, gmm, rowmax, out);
}